// RAG4DyG_45612552683762
// MI455X (gfx1250) — compile-verified
//
#include <hip/hip_runtime.h>

typedef float v2f __attribute__((ext_vector_type(2)));
typedef float v8f __attribute__((ext_vector_type(8)));

#define FEAT   64
#define DIM    64
#define BATCH  256
#define L_NEI  50
#define T_DIM  16
#define K_RET  8
#define SEQ    50

// ---------------------------------------------------------------------------
// f32 WMMA 16x16x4 tile helper.
//   A (16xK, f32) staged in LDS row-major, lda even (ds_load_b64 fragments):
//     lane<16 : v0=A[m][4s+0], v1=A[m][4s+1];  lane>=16: K+2/K+3   (ISA 7.12.2)
//   B (Kx16, f32) read from global row-major W[k*ldw + n] (mirrored layout).
//   C frag: lane holds column n=c0+(lane&15), rows m=r+8*(lane>>4).
// ---------------------------------------------------------------------------
__device__ __forceinline__ v8f wmma_tile(const float* __restrict__ sA, int lda,
                                         const float* __restrict__ W, int ldw,
                                         int c0, int ksteps, v8f acc, int lane) {
  const int half = lane >> 4;
  const int l16  = lane & 15;
  const int n    = c0 + l16;
#pragma unroll 4
  for (int s = 0; s < ksteps; ++s) {
    const int k = 4 * s + 2 * half;
    v2f a = *(const v2f*)(sA + l16 * lda + k);
    v2f b;
    b.x = W[k * ldw + n];
    b.y = W[(k + 1) * ldw + n];
    acc = __builtin_amdgcn_wmma_f32_16x16x4_f32(false, a, false, b,
                                                (short)0, acc, false, false);
  }
  return acc;
}

__device__ __forceinline__ float wred32(float x) {
#pragma unroll
  for (int m = 16; m >= 1; m >>= 1) x += __shfl_xor(x, m, 32);
  return x;
}

// ---------------------------------------------------------------------------
// feats(): gather node row + cos(time*w) -> [16 rows x 80] @ [80x64] WMMA,
// add rank-1 edge/co/skill terms + biases, LayerNorm, write [rows,64].
// grid = 12800/16 blocks, 128 threads (4 waves).
// ---------------------------------------------------------------------------
__global__ __launch_bounds__(128) void k_feats(
    const float* __restrict__ node_raw, const float* __restrict__ edge_raw,
    const int* __restrict__ nbr, const int* __restrict__ eids,
    const float* __restrict__ times, const int* __restrict__ cur_ids,
    const float* __restrict__ w_time,
    const float* __restrict__ Wf, const float* __restrict__ bf,
    const float* __restrict__ We, const float* __restrict__ be,
    const float* __restrict__ Wt, const float* __restrict__ bt,
    const float* __restrict__ Ws, const float* __restrict__ bs,
    const float* __restrict__ lng, const float* __restrict__ lnb,
    float* __restrict__ out_h) {
  __shared__ __align__(16) float sA[16][88];   // 64 node feats + 16 cos
  __shared__ float sOut[16][72];
  __shared__ float sE0[16];
  __shared__ float sCS[16];

  const int tid = threadIdx.x;
  const int R0  = blockIdx.x * 16;
  const int g   = tid & 7, i = tid >> 3;
  const int row = R0 + i;
  const int b   = row / L_NEI;
  const int nid = nbr[row];

  const float4* s4 = (const float4*)(node_raw + (long)nid * FEAT + g * 8);
  *(float4*)&sA[i][g * 8]     = s4[0];
  *(float4*)&sA[i][g * 8 + 4] = s4[1];
  if (g == 0) {
    float tm = times[row];
#pragma unroll
    for (int k = 0; k < T_DIM; ++k) sA[i][64 + k] = cosf(tm * w_time[k]);
  } else if (g == 1) {
    sE0[i] = edge_raw[(long)eids[row] * 4];
  } else if (g == 2) {
    int cd    = cur_ids[b];
    int skill = (int)node_raw[(long)cd * FEAT];
    int sh    = (int)node_raw[(long)nid * FEAT];
    sCS[i] = (nid == cd ? 1.f : 0.f) + (sh == skill ? 1.f : 0.f);
  }
  __syncthreads();

  const int wv = tid >> 5, lane = tid & 31;
  const int half = lane >> 4, l16 = lane & 15;
  const int c0 = wv * 16, n = c0 + l16;
  v8f acc = {};
  acc = wmma_tile(&sA[0][0], 88, Wf, 64, c0, 16, acc, lane);   // K 0..63
  acc = wmma_tile(&sA[0][64], 88, Wt, 64, c0, 4, acc, lane);   // K 64..79
  const float bias = bf[n] + be[n] + bt[n] + 2.f * bs[n];
  const float wen = We[n], wsn = Ws[n];
#pragma unroll
  for (int r = 0; r < 8; ++r) {
    int m = r + 8 * half;
    sOut[m][n] = acc[r] + bias + sE0[m] * wen + sCS[m] * wsn;
  }
  __syncthreads();

  for (int mr = 4 * wv; mr < 4 * wv + 4; ++mr) {
    float x0 = sOut[mr][lane], x1 = sOut[mr][lane + 32];
    float mu = wred32(x0 + x1) * (1.f / 64.f);
    float d0 = x0 - mu, d1 = x1 - mu;
    float var = wred32(d0 * d0 + d1 * d1) * (1.f / 64.f);
    float rs = rsqrtf(var + 1e-5f);
    long base = (long)(R0 + mr) * DIM;
    out_h[base + lane]      = d0 * rs * lng[lane]      + lnb[lane];
    out_h[base + lane + 32] = d1 * rs * lng[lane + 32] + lnb[lane + 32];
  }
}

// Hp = node_raw @ Wf + bf   (4000x64 @ 64x64), grid 250 x 128 threads.
__global__ __launch_bounds__(128) void k_hp(
    const float* __restrict__ node_raw, const float* __restrict__ Wf,
    const float* __restrict__ bf, float* __restrict__ Hp) {
  __shared__ __align__(16) float sA[16][72];
  const int tid = threadIdx.x, R0 = blockIdx.x * 16;
  const int g = tid & 7, i = tid >> 3;
  const float4* s4 = (const float4*)(node_raw + (long)(R0 + i) * FEAT + g * 8);
  *(float4*)&sA[i][g * 8]     = s4[0];
  *(float4*)&sA[i][g * 8 + 4] = s4[1];
  __syncthreads();
  const int wv = tid >> 5, lane = tid & 31;
  const int half = lane >> 4, l16 = lane & 15;
  const int c0 = wv * 16, n = c0 + l16;
  v8f acc = {};
  acc = wmma_tile(&sA[0][0], 72, Wf, 64, c0, 16, acc, lane);
#pragma unroll
  for (int r = 0; r < 8; ++r) {
    int m = r + 8 * half;
    Hp[(long)(R0 + m) * DIM + n] = acc[r] + bf[n];
  }
}

// ---------------------------------------------------------------------------
// GRU: 16 batch rows per workgroup, h resident in LDS for all 50 steps.
// Per step: gi = x_t@Wih, gh = h@Whh via WMMA (12 N-tiles of 16, K=64),
// then elementwise gates. grid 16 x 128 threads.
// ---------------------------------------------------------------------------
__global__ __launch_bounds__(128) void k_gru(
    const float* __restrict__ x_all,   // [256*50*64]
    const float* __restrict__ Wih, const float* __restrict__ Whh,
    const float* __restrict__ bih, const float* __restrict__ bhh,
    float* __restrict__ h_out) {
  __shared__ __align__(16) float sx[16][72];
  __shared__ __align__(16) float sh[16][72];
  __shared__ float sgi[16][200];
  __shared__ float sgh[16][200];
  const int tid = threadIdx.x, B0 = blockIdx.x * 16;
  for (int idx = tid; idx < 16 * 72; idx += 128) ((float*)sh)[idx] = 0.f;
  const int g = tid & 7, i = tid >> 3;
  const int wv = tid >> 5, lane = tid & 31;
  const int half = lane >> 4, l16 = lane & 15;
  __syncthreads();

  for (int t = 0; t < SEQ; ++t) {
    const float4* s4 =
        (const float4*)(x_all + ((long)(B0 + i) * SEQ + t) * DIM + g * 8);
    *(float4*)&sx[i][g * 8]     = s4[0];
    *(float4*)&sx[i][g * 8 + 4] = s4[1];
    __syncthreads();
    for (int nt = wv; nt < 12; nt += 4) {
      const int c0 = nt * 16, n = c0 + l16;
      v8f ai = {}; ai = wmma_tile(&sx[0][0], 72, Wih, 192, c0, 16, ai, lane);
      v8f ah = {}; ah = wmma_tile(&sh[0][0], 72, Whh, 192, c0, 16, ah, lane);
#pragma unroll
      for (int r = 0; r < 8; ++r) {
        int m = r + 8 * half;
        sgi[m][n] = ai[r];
        sgh[m][n] = ah[r];
      }
    }
    __syncthreads();
#pragma unroll
    for (int u = 0; u < 8; ++u) {
      int idx = tid + 128 * u, m = idx >> 6, j = idx & 63;
      float rr = 1.f / (1.f + expf(-(sgi[m][j] + bih[j] + sgh[m][j] + bhh[j])));
      float zz = 1.f / (1.f + expf(-(sgi[m][64 + j] + bih[64 + j] +
                                     sgh[m][64 + j] + bhh[64 + j])));
      float nn = tanhf(sgi[m][128 + j] + bih[128 + j] +
                       rr * (sgh[m][128 + j] + bhh[128 + j]));
      sh[m][j] = (1.f - zz) * nn + zz * sh[m][j];
    }
    __syncthreads();
  }
#pragma unroll
  for (int u = 0; u < 8; ++u) {
    int idx = tid + 128 * u, m = idx >> 6, j = idx & 63;
    h_out[(long)(B0 + m) * DIM + j] = sh[m][j];
  }
}

// ---------------------------------------------------------------------------
// GCN pool: one workgroup (256 thr, 8 waves) per sample. Deterministic
// LDS dedup/edges, per-node deterministic agg, WMMA transform+relu+mean.
// ---------------------------------------------------------------------------
__global__ __launch_bounds__(256) void k_gcn(
    const float* __restrict__ Hp, const int* __restrict__ pool,
    const int* __restrict__ ridx, const float* __restrict__ Wg,
    const float* __restrict__ bg, float* __restrict__ ctx) {
  __shared__ int demoIds[400];
  __shared__ int hkey[1024];
  __shared__ int hval[1024];
  __shared__ int cids[400];
  __shared__ float deg[400];
  __shared__ float dinv[400];
  __shared__ int eS[400], eD[400];
  __shared__ __align__(16) float agg[128][66];
  __shared__ float pooled[64];
  __shared__ int nactS;

  const int tid = threadIdx.x, b = blockIdx.x;
  for (int h = tid; h < 1024; h += 256) hkey[h] = 0;
  for (int e = tid; e < 400; e += 256) {
    int k = e / SEQ, j = e % SEQ;
    demoIds[e] = pool[(long)ridx[b * K_RET + k] * SEQ + j];
  }
  __syncthreads();

  if (tid == 0) {                     // serial insert => deterministic order
    int c = 0;
    for (int e = 0; e < 400; ++e) {
      int id = demoIds[e];
      if (id <= 0) continue;
      unsigned h = ((unsigned)id * 2654435761u >> 16) & 1023u;
      for (;;) {
        int kk = hkey[h];
        if (kk == id) break;
        if (kk == 0) { hkey[h] = id; hval[h] = c; cids[c] = id; ++c; break; }
        h = (h + 1) & 1023u;
      }
    }
    nactS = c;
  }
  __syncthreads();
  const int nact = nactS;

  if (tid < 8) {                      // path edges per demo sequence
    int k = tid, nv = -1;
    for (int j = SEQ - 1; j >= 0; --j) {
      int id = demoIds[k * SEQ + j], slot = k * SEQ + j;
      if (id > 0) {
        unsigned h = ((unsigned)id * 2654435761u >> 16) & 1023u;
        while (hkey[h] != id) h = (h + 1) & 1023u;
        int c = hval[h];
        if (nv >= 0) { eS[slot] = c; eD[slot] = nv; } else eS[slot] = -1;
        nv = c;
      } else eS[slot] = -1;
    }
  }
  for (int c = tid; c < nact; c += 256) deg[c] = 1.0f;   // self loop + act
  __syncthreads();
  for (int e = tid; e < 400; e += 256)
    if (eS[e] >= 0) atomicAdd(&deg[eD[e]], 1.0f);        // exact integer adds
  __syncthreads();
  for (int c = tid; c < nact; c += 256) dinv[c] = rsqrtf(deg[c]);
  __syncthreads();

  const int wid = tid >> 5, lane = tid & 31;
  const int half = lane >> 4, l16 = lane & 15;
  float pPart = 0.f;
  for (int cb = 0; cb < nact; cb += 128) {
    const int rows = min(128, nact - cb);
    for (int c = cb + wid; c < cb + rows; c += 8) {     // warp per node
      int gid = cids[c];
      float di = dinv[c], d2 = di * di;
      const float* hp = Hp + (long)gid * DIM;
      float a0 = d2 * hp[lane], a1 = d2 * hp[lane + 32];
      for (int e = 0; e < 400; ++e) {                    // fixed-order scan
        int s = eS[e];
        if (s >= 0 && eD[e] == c) {
          float w = dinv[s] * di;
          const float* hs = Hp + (long)cids[s] * DIM;
          a0 += w * hs[lane];
          a1 += w * hs[lane + 32];
        }
      }
      agg[c - cb][lane] = a0;
      agg[c - cb][lane + 32] = a1;
    }
    const int padded = (rows + 15) & ~15;
    for (int idx = tid; idx < (padded - rows) * 64; idx += 256)
      agg[rows + (idx >> 6)][idx & 63] = 0.f;
    __syncthreads();
    if (wid < 4) {                                       // WMMA transform+pool
      const int c0 = wid * 16, n = c0 + l16;
      const float bn = bg[n];
      for (int mt = 0; mt < padded / 16; ++mt) {
        v8f acc = {};
        acc = wmma_tile(&agg[mt * 16][0], 66, Wg, 64, c0, 16, acc, lane);
#pragma unroll
        for (int r = 0; r < 8; ++r) {
          int m = cb + mt * 16 + r + 8 * half;
          if (m < nact) pPart += fmaxf(acc[r] + bn, 0.f);
        }
      }
    }
    __syncthreads();
  }
  pPart += __shfl_xor(pPart, 16, 32);
  if (wid < 4 && lane < 16) pooled[wid * 16 + lane] = pPart;
  __syncthreads();
  if (tid < 64) {
    float v = (nact > 0) ? pooled[tid] / (float)nact : fmaxf(bg[tid], 0.f);
    ctx[(long)b * DIM + tid] = v;
  }
}

// fused = relu([sl, ctx] @ Wmlp + bmlp); out1 = fused@Wout+bout; out2 = qn@Wout+bout
__global__ __launch_bounds__(128) void k_head(
    const float* __restrict__ sl, const float* __restrict__ ctxv,
    const float* __restrict__ qn,
    const float* __restrict__ Wm, const float* __restrict__ bm,
    const float* __restrict__ Wo, const float* __restrict__ bo,
    float* __restrict__ out) {
  __shared__ __align__(16) float sC[16][136];
  __shared__ __align__(16) float sF[16][72];
  __shared__ __align__(16) float sQ[16][72];
  const int tid = threadIdx.x, R0 = blockIdx.x * 16;
  const int g = tid & 7, i = tid >> 3;
  const long base = (long)(R0 + i) * DIM + g * 8;
  *(float4*)&sC[i][g * 8]          = *(const float4*)(sl + base);
  *(float4*)&sC[i][g * 8 + 4]      = *(const float4*)(sl + base + 4);
  *(float4*)&sC[i][64 + g * 8]     = *(const float4*)(ctxv + base);
  *(float4*)&sC[i][64 + g * 8 + 4] = *(const float4*)(ctxv + base + 4);
  *(float4*)&sQ[i][g * 8]          = *(const float4*)(qn + base);
  *(float4*)&sQ[i][g * 8 + 4]      = *(const float4*)(qn + base + 4);
  __syncthreads();
  const int wv = tid >> 5, lane = tid & 31;
  const int half = lane >> 4, l16 = lane & 15;
  const int c0 = wv * 16, n = c0 + l16;
  v8f aF = {};
  aF = wmma_tile(&sC[0][0], 136, Wm, 64, c0, 32, aF, lane);
#pragma unroll
  for (int r = 0; r < 8; ++r) {
    int m = r + 8 * half;
    sF[m][n] = fmaxf(aF[r] + bm[n], 0.f);
  }
  __syncthreads();
  v8f a1 = {}; a1 = wmma_tile(&sF[0][0], 72, Wo, 64, c0, 16, a1, lane);
  v8f a2 = {}; a2 = wmma_tile(&sQ[0][0], 72, Wo, 64, c0, 16, a2, lane);
#pragma unroll
  for (int r = 0; r < 8; ++r) {
    int m = r + 8 * half;
    out[(long)(R0 + m) * DIM + n]                       = a1[r] + bo[n];
    out[(long)BATCH * DIM + (long)(R0 + m) * DIM + n]   = a2[r] + bo[n];
  }
}

extern "C" void kernel_launch(void* const* d_in, const int* in_sizes, int n_in,
                              void* d_out, int out_size, void* d_ws,
                              size_t ws_size, hipStream_t stream) {
  (void)in_sizes; (void)n_in; (void)out_size; (void)ws_size;
  const float* node_raw = (const float*)d_in[0];
  const float* edge_raw = (const float*)d_in[1];
  const int*   pool     = (const int*)d_in[2];
  const int*   ridx     = (const int*)d_in[3];
  const int*   src_ids  = (const int*)d_in[4];
  const int*   dst_ids  = (const int*)d_in[5];
  const int*   lnbr     = (const int*)d_in[6];
  const int*   leid     = (const int*)d_in[7];
  const float* ltim     = (const float*)d_in[8];
  const int*   dnbr     = (const int*)d_in[9];
  const int*   deid     = (const int*)d_in[10];
  const float* dtim     = (const float*)d_in[11];
  const float* w_time   = (const float*)d_in[12];
  const float *Wf = (const float*)d_in[13], *bf = (const float*)d_in[14];
  const float *We = (const float*)d_in[15], *be = (const float*)d_in[16];
  const float *Wt = (const float*)d_in[17], *bt = (const float*)d_in[18];
  const float *Ws = (const float*)d_in[19], *bs = (const float*)d_in[20];
  const float *lng = (const float*)d_in[21], *lnb = (const float*)d_in[22];
  const float *g1Wih = (const float*)d_in[23], *g1Whh = (const float*)d_in[24];
  const float *g1bih = (const float*)d_in[25], *g1bhh = (const float*)d_in[26];
  const float *g2Wih = (const float*)d_in[27], *g2Whh = (const float*)d_in[28];
  const float *g2bih = (const float*)d_in[29], *g2bhh = (const float*)d_in[30];
  const float *Wg = (const float*)d_in[31], *bg = (const float*)d_in[32];
  const float *Wm = (const float*)d_in[33], *bm = (const float*)d_in[34];
  const float *Wo = (const float*)d_in[35], *bo = (const float*)d_in[36];

  float* ws      = (float*)d_ws;
  float* local_h = ws;                                // 256*50*64
  float* dst_h   = local_h + BATCH * SEQ * DIM;       // 256*50*64
  float* Hp      = dst_h + BATCH * SEQ * DIM;         // 4000*64
  float* sl      = Hp + 4000 * DIM;                   // 256*64
  float* qn      = sl + BATCH * DIM;                  // 256*64
  float* ctx     = qn + BATCH * DIM;                  // 256*64

  k_feats<<<(BATCH * L_NEI) / 16, 128, 0, stream>>>(
      node_raw, edge_raw, lnbr, leid, ltim, dst_ids, w_time,
      Wf, bf, We, be, Wt, bt, Ws, bs, lng, lnb, local_h);
  k_feats<<<(BATCH * L_NEI) / 16, 128, 0, stream>>>(
      node_raw, edge_raw, dnbr, deid, dtim, src_ids, w_time,
      Wf, bf, We, be, Wt, bt, Ws, bs, lng, lnb, dst_h);
  k_hp<<<4000 / 16, 128, 0, stream>>>(node_raw, Wf, bf, Hp);
  k_gru<<<BATCH / 16, 128, 0, stream>>>(local_h, g1Wih, g1Whh, g1bih, g1bhh, sl);
  k_gru<<<BATCH / 16, 128, 0, stream>>>(dst_h, g2Wih, g2Whh, g2bih, g2bhh, qn);
  k_gcn<<<BATCH, 256, 0, stream>>>(Hp, pool, ridx, Wg, bg, ctx);
  k_head<<<BATCH / 16, 128, 0, stream>>>(sl, ctx, qn, Wm, bm, Wo, bo,
                                         (float*)d_out);
}